// CapsuleNetwork_71906342469702
// MI455X (gfx1250) — compile-verified
//
#include <hip/hip_runtime.h>

#define B_  1024
#define S_  200
#define K_  4
#define D_  64
#define KD_ 256   // K_*D_
#define CH_ 8     // s-positions per routing chunk
#define NCH_ (S_ / CH_)   // 25

typedef __attribute__((ext_vector_type(2))) float v2f;
typedef __attribute__((ext_vector_type(8))) float v8f;

// ---------------------------------------------------------------------------
// Kernel 1: hat[b, k, s, d] = sum_d' X[b, s, d'] * W[s, k*64+d, d']
// One block = one s, 128 batch rows. 256 threads = 8 wave32 waves.
// Each wave owns a 16-row M tile; loops 16 N-tiles x 16 K-steps of
// v_wmma_f32_16x16x4_f32, accumulating in an 8-VGPR f32 tile.
// ---------------------------------------------------------------------------
__global__ __launch_bounds__(256) void capsule_hat_gemm(
    const float* __restrict__ x,    // [B, S, D]
    const float* __restrict__ w,    // [1, S, K*D, D]
    float* __restrict__ hat)        // [B, K, S, D]  (workspace)
{
    extern __shared__ float lds[];
    float* As = lds;                 // 128 x 64, row stride 65 (bank padding)
    float* Ws = lds + 128 * 65;      // 256 x 64, row stride 65

    const int s   = blockIdx.y;
    const int b0  = blockIdx.x * 128;
    const int tid = threadIdx.x;

    // Stage X tile: rows are 256B contiguous chunks -> coalesced.
    for (int i = tid; i < 128 * 64; i += 256) {
        const int row = i >> 6, d = i & 63;
        As[row * 65 + d] = x[(size_t)(b0 + row) * (S_ * D_) + (size_t)s * D_ + d];
    }
    // Stage W[s]: one contiguous 64KB chunk -> fully coalesced.
    const float* wsrc = w + (size_t)s * (KD_ * D_);
    for (int i = tid; i < KD_ * 64; i += 256) {
        const int row = i >> 6, d = i & 63;
        Ws[row * 65 + d] = wsrc[i];
    }
    __syncthreads();

    const int wave = tid >> 5;       // 0..7 -> M tile
    const int lane = tid & 31;
    const int lrow = lane & 15;
    const int grp  = lane >> 4;      // 0: lanes 0-15, 1: lanes 16-31
    const int m0   = wave * 16;

    for (int nt = 0; nt < 16; ++nt) {
        v8f acc = {};
        #pragma unroll
        for (int kk = 0; kk < 16; ++kk) {
            const int kb = kk * 4;
            v2f a, bf;
            // A (16x4, M x K): a[v] = A[lrow][kb + v + 2*grp]
            a.x = As[(m0 + lrow) * 65 + kb + 0 + 2 * grp];
            a.y = As[(m0 + lrow) * 65 + kb + 1 + 2 * grp];
            // B (4x16, K x N): B[k][n] = W[s][nt*16+n][kb+k]
            bf.x = Ws[(nt * 16 + lrow) * 65 + kb + 0 + 2 * grp];
            bf.y = Ws[(nt * 16 + lrow) * 65 + kb + 1 + 2 * grp];
            acc = __builtin_amdgcn_wmma_f32_16x16x4_f32(
                false, a, false, bf, (short)0, acc, false, false);
        }
        // D tile (16x16 f32): lane -> n = lane&15; VGPR v -> m = v + 8*grp.
        const int ncol = nt * 16 + lrow;
        const int kcap = ncol >> 6;     // which capsule
        const int d    = ncol & 63;     // position inside capsule
        #pragma unroll
        for (int v = 0; v < 8; ++v) {
            const int b = b0 + m0 + v + 8 * grp;
            hat[(((size_t)b * K_ + kcap) * S_ + s) * D_ + d] = acc[v];
        }
    }
}

// ---------------------------------------------------------------------------
// Async staging helper: bring hat[b, 0..3, s0..s0+7, 0..63] (8 KB) into LDS
// with GLOBAL_LOAD_ASYNC_TO_LDS_B128 (ASYNCcnt-tracked). 512 x 16B over 256
// threads = 2 instructions per wave per chunk; completion is in-order.
// ---------------------------------------------------------------------------
__device__ __forceinline__ void async_load_chunk(
    const float* __restrict__ hb, float* dstLds, int s0, int tid)
{
    const unsigned long long gbase = (unsigned long long)(uintptr_t)hb;
    #pragma unroll
    for (int j = 0; j < 2; ++j) {
        const int slot = tid + 256 * j;          // 0..511  (vec4 slots)
        const int kk = slot >> 7;                // 0..3
        const int sc = (slot >> 4) & 7;          // 0..7
        const int d4 = (slot & 15) << 2;         // 0,4,...,60
        const unsigned goff  = (unsigned)((((kk * S_ + s0 + sc) * D_) + d4) * sizeof(float));
        // low 32 bits of a generic LDS pointer == offset in the LDS aperture
        const unsigned laddr = (unsigned)(uintptr_t)(dstLds + ((kk * CH_ + sc) * D_ + d4));
        asm volatile("global_load_async_to_lds_b128 %0, %1, %2 offset:0"
                     :: "v"(laddr), "v"(goff), "s"(gbase) : "memory");
    }
}

// ---------------------------------------------------------------------------
// Kernel 2: fused dynamic routing, 3 sweeps over hat (instead of 5).
// Per chunk of 8 positions: delta(prev cap) -> cw update -> softmax -> sw ->
// accumulate new cap, all from one LDS-resident copy of the chunk, which is
// double-buffered and fetched with async LDS loads one chunk ahead.
// One block per batch row; 256 threads = 8 wave32 waves.
// ---------------------------------------------------------------------------
__global__ __launch_bounds__(256) void capsule_routing_fused(
    const float* __restrict__ hat,      // [B, K, S, D]
    const float* __restrict__ item_eb,  // [B, D]
    const int*   __restrict__ mask,     // [B, S]
    float* __restrict__ out)            // [B*K*D] ++ [B*D]
{
    __shared__ __align__(16) float hbuf[2][K_][CH_][D_];  // 2 x 8 KB
    __shared__ float cw[K_][S_];
    __shared__ float swb[K_][CH_];
    __shared__ float capOld[K_][D_];
    __shared__ float capNew[K_][D_];
    __shared__ float scale[K_];
    __shared__ float att[K_];
    __shared__ int   widx;
    __shared__ float validf[S_];

    const int b    = blockIdx.x;
    const int tid  = threadIdx.x;
    const int k    = tid >> 6;     // 0..3
    const int d    = tid & 63;     // 0..63
    const int wave = tid >> 5;     // 0..7
    const int lane = tid & 31;

    for (int si = tid; si < S_; si += 256) {
        validf[si] = (mask[b * S_ + si] != 0) ? 1.f : 0.f;
        cw[0][si] = 0.f; cw[1][si] = 0.f; cw[2][si] = 0.f; cw[3][si] = 0.f;
    }
    __syncthreads();

    const float* hb = hat + (size_t)b * K_ * S_ * D_;

    for (int sweep = 0; sweep < 3; ++sweep) {
        const bool first = (sweep == 0);
        float acc = 0.f;

        async_load_chunk(hb, &hbuf[0][0][0][0], 0, tid);

        for (int c = 0; c < NCH_; ++c) {
            const int bufc = c & 1;
            if (c + 1 < NCH_) {
                async_load_chunk(hb, &hbuf[(c + 1) & 1][0][0][0], (c + 1) * CH_, tid);
                asm volatile("s_wait_asynccnt 0x2" ::: "memory");  // chunk c done
            } else {
                asm volatile("s_wait_asynccnt 0x0" ::: "memory");
            }
            __syncthreads();                 // chunk c visible to all waves
            const int s0 = c * CH_;

            if (!first) {
                // delta[k][s] = <hat[b,k,s,:], capOld[k,:]> ; 32 (k,sc) pairs
                for (int p = wave; p < K_ * CH_; p += 8) {
                    const int kk = p & 3;
                    const int sc = p >> 2;
                    float v = fmaf(hbuf[bufc][kk][sc][lane], capOld[kk][lane],
                                   hbuf[bufc][kk][sc][lane + 32] * capOld[kk][lane + 32]);
                    #pragma unroll
                    for (int off = 16; off > 0; off >>= 1)
                        v += __shfl_xor(v, off, 32);       // wave32 reduce
                    if (lane == 0) cw[kk][s0 + sc] += v;
                }
                __syncthreads();
                // softmax over K per position (local in s), then mask
                if (tid < CH_) {
                    const int s = s0 + tid;
                    const float c0 = cw[0][s], c1 = cw[1][s], c2 = cw[2][s], c3 = cw[3][s];
                    const float mx = fmaxf(fmaxf(c0, c1), fmaxf(c2, c3));
                    const float e0 = expf(c0 - mx), e1 = expf(c1 - mx);
                    const float e2 = expf(c2 - mx), e3 = expf(c3 - mx);
                    const float inv = validf[s] / (e0 + e1 + e2 + e3);
                    swb[0][tid] = e0 * inv; swb[1][tid] = e1 * inv;
                    swb[2][tid] = e2 * inv; swb[3][tid] = e3 * inv;
                }
                __syncthreads();
            }

            // accumulate cap (thread owns (k,d); reads are LDS-local)
            if (first) {
                #pragma unroll
                for (int sc = 0; sc < CH_; ++sc)
                    acc = fmaf(0.25f * validf[s0 + sc], hbuf[bufc][k][sc][d], acc);
            } else {
                #pragma unroll
                for (int sc = 0; sc < CH_; ++sc)
                    acc = fmaf(swb[k][sc], hbuf[bufc][k][sc][d], acc);
            }
            __syncthreads();   // protect hbuf[bufc] before it is re-targeted
        }

        // squash: scale = n/(1+n)/sqrt(n+1e-9)
        capNew[k][d] = acc;
        __syncthreads();
        if (tid < K_) {
            float n = 0.f;
            for (int dd = 0; dd < D_; ++dd) {
                const float c = capNew[tid][dd];
                n = fmaf(c, c, n);
            }
            scale[tid] = n / (1.f + n) * rsqrtf(n + 1e-9f);
        }
        __syncthreads();
        capOld[k][d] = capNew[k][d] * scale[k];   // becomes "previous cap"
        __syncthreads();
    }

    // --- outputs ---
    out[((size_t)b * K_ + k) * D_ + d] = capOld[k][d];   // interest_capsule

    if (tid < K_) {
        float a = 0.f;
        for (int dd = 0; dd < D_; ++dd)
            a = fmaf(capOld[tid][dd], item_eb[b * D_ + dd], a);
        att[tid] = a;   // softmax is monotonic -> argmax on logits is identical
    }
    __syncthreads();
    if (tid == 0) {
        int best = 0; float bv = att[0];
        for (int kk = 1; kk < K_; ++kk)
            if (att[kk] > bv) { bv = att[kk]; best = kk; }  // first-max ties
        widx = best;
    }
    __syncthreads();
    if (tid < D_)
        out[(size_t)B_ * K_ * D_ + (size_t)b * D_ + tid] = capOld[widx][tid];
}

// ---------------------------------------------------------------------------
extern "C" void kernel_launch(void* const* d_in, const int* in_sizes, int n_in,
                              void* d_out, int out_size, void* d_ws, size_t ws_size,
                              hipStream_t stream) {
    (void)in_sizes; (void)n_in; (void)out_size; (void)ws_size;
    const float* x        = (const float*)d_in[0];   // item_his_emb [B,S,D]
    const float* item_eb  = (const float*)d_in[1];   // [B,D]
    const int*   mask     = (const int*)d_in[2];     // [B,S]
    const float* w        = (const float*)d_in[3];   // [1,S,K*D,D]
    float*       out      = (float*)d_out;           // [B*K*D] ++ [B*D]
    float*       hat      = (float*)d_ws;            // [B,K,S,D] = 210 MB

    const size_t lds_bytes = (size_t)(128 * 65 + 256 * 65) * sizeof(float); // ~97.5 KB
    dim3 g1(B_ / 128, S_);
    capsule_hat_gemm<<<g1, 256, lds_bytes, stream>>>(x, w, hat);
    capsule_routing_fused<<<B_, 256, 0, stream>>>(hat, item_eb, mask, out);
}